// GaussianModel_36163624633074
// MI455X (gfx1250) — compile-verified
//
#include <hip/hip_runtime.h>
#include <hip/hip_bf16.h>

typedef float v2f __attribute__((ext_vector_type(2)));
typedef float v8f __attribute__((ext_vector_type(8)));

#define NPTS      1000000
#define KREND     1000
#define IMG_H     256
#define IMG_W     256
#define NPIX      (IMG_H * IMG_W)
#define CAND_MAX  4096
#define HIST_BINS 65536

#define ALIGN256(x) ((((size_t)(x)) + 255) & ~((size_t)255))

__device__ __forceinline__ float sigmoidf_(float t) {
    return 1.0f / (1.0f + __expf(-t));
}

// cross-half (lane ^ 16) swap on a wave32: VALU v_permlanex16_b32, no LDS pipe
__device__ __forceinline__ float halfswap_f32(float v) {
#if __has_builtin(__builtin_amdgcn_permlanex16)
    int r = __builtin_amdgcn_permlanex16(__float_as_int(v), __float_as_int(v),
                                         0x76543210, 0xFEDCBA98, false, false);
    return __int_as_float(r);
#else
    return __shfl_xor(v, 16, 32);
#endif
}

// ---- 1) sortable depth keys (monotone uint mapping of float depth) ----
__global__ void depth_keys_kernel(const float* __restrict__ pos,
                                  const float* __restrict__ cam,
                                  unsigned* __restrict__ keys, int n) {
    int i = blockIdx.x * blockDim.x + threadIdx.x;
    if (i >= n) return;
    __builtin_prefetch(&pos[3 * i + 768], 0, 1);   // global_prefetch_b8
    float x = pos[3 * i], y = pos[3 * i + 1], z = pos[3 * i + 2];
    float d = cam[8] * x + cam[9] * y + cam[10] * z + cam[11];
    unsigned b = __float_as_uint(d);
    keys[i] = (b & 0x80000000u) ? ~b : (b | 0x80000000u);
}

// ---- 2) zero histogram + counters ----
__global__ void zero_kernel(unsigned* __restrict__ hist) {
    int i = blockIdx.x * blockDim.x + threadIdx.x;
    if (i < HIST_BINS + 64) hist[i] = 0u;
}

// ---- 3) 16-bit histogram ----
__global__ void hist_kernel(const unsigned* __restrict__ keys,
                            unsigned* __restrict__ hist, int n) {
    int i = blockIdx.x * blockDim.x + threadIdx.x;
    if (i >= n) return;
    atomicAdd(&hist[keys[i] >> 16], 1u);
}

// ---- 4) find threshold bin so that count(key>>16 >= t) >= K ----
__global__ void thresh_kernel(const unsigned* __restrict__ hist,
                              unsigned* __restrict__ ctr, int K) {
    if (blockIdx.x == 0 && threadIdx.x == 0) {
        unsigned cum = 0; int t = 0;
        for (int b = HIST_BINS - 1; b >= 0; --b) {
            cum += hist[b];
            if (cum >= (unsigned)K) { t = b; break; }
        }
        ctr[1] = (unsigned)t;
    }
}

// ---- 5) compact candidates: pack (key, ~idx) into u64 for stable sort ----
__global__ void compact_kernel(const unsigned* __restrict__ keys,
                               unsigned* __restrict__ ctr,
                               unsigned long long* __restrict__ cand, int n) {
    int i = blockIdx.x * blockDim.x + threadIdx.x;
    if (i >= n) return;
    unsigned k = keys[i];
    if ((k >> 16) >= ctr[1]) {
        unsigned p = atomicAdd(&ctr[0], 1u);
        if (p < (unsigned)CAND_MAX)
            cand[p] = (((unsigned long long)k) << 32) |
                      (unsigned long long)(~(unsigned)i);
    }
}

// ---- 6) single-workgroup LDS bitonic sort (descending), emit top-K ----
__global__ __launch_bounds__(512) void sort_kernel(
        const unsigned long long* __restrict__ cand,
        const unsigned* __restrict__ ctr,
        unsigned* __restrict__ selIdx, int K) {
    __shared__ unsigned long long s[CAND_MAX];
    unsigned count = ctr[0];
    if (count > (unsigned)CAND_MAX) count = CAND_MAX;
    for (int i = threadIdx.x; i < CAND_MAX; i += blockDim.x)
        s[i] = (i < (int)count) ? cand[i] : 0ull;
    __syncthreads();
    for (unsigned k = 2; k <= CAND_MAX; k <<= 1) {
        for (unsigned j = k >> 1; j > 0; j >>= 1) {
            for (unsigned i = threadIdx.x; i < CAND_MAX; i += blockDim.x) {
                unsigned ixj = i ^ j;
                if (ixj > i) {
                    unsigned long long a = s[i], b = s[ixj];
                    bool up = ((i & k) == 0);          // descending segment
                    bool sw = up ? (a < b) : (a > b);
                    if (sw) { s[i] = b; s[ixj] = a; }
                }
            }
            __syncthreads();
        }
    }
    for (int i = threadIdx.x; i < K; i += blockDim.x)
        selIdx[i] = ~(unsigned)(s[i] & 0xFFFFFFFFull);
}

// ---- 7) full preprocessing for the K selected gaussians only ----
// geom:  per gaussian 8 floats {mu.x, mu.y, cA, cB, cC, op, 0, 0}
// colp:  per gaussian 4 floats {r, g, b, 0}   (pad slot gives B=0 for lanes n>=3)
__global__ void prep_kernel(const float* __restrict__ pos,
                            const float* __restrict__ scl,
                            const float* __restrict__ rot,
                            const float* __restrict__ col,
                            const float* __restrict__ opa,
                            const float* __restrict__ sh_c,
                            const float* __restrict__ cam,
                            const float* __restrict__ campos,
                            const unsigned* __restrict__ selIdx,
                            float* __restrict__ geom,
                            float* __restrict__ colp, int K) {
    int k = blockIdx.x * blockDim.x + threadIdx.x;
    if (k >= K) return;
    int i = (int)selIdx[k];

    float x = pos[3 * i], y = pos[3 * i + 1], z = pos[3 * i + 2];

    // --- quaternion -> R ---
    float qw = rot[4 * i], qx = rot[4 * i + 1], qy = rot[4 * i + 2], qz = rot[4 * i + 3];
    float rn = rsqrtf(qw * qw + qx * qx + qy * qy + qz * qz);
    qw *= rn; qx *= rn; qy *= rn; qz *= rn;
    float R00 = 1.f - 2.f * (qy * qy + qz * qz), R01 = 2.f * (qx * qy - qw * qz), R02 = 2.f * (qx * qz + qw * qy);
    float R10 = 2.f * (qx * qy + qw * qz), R11 = 1.f - 2.f * (qx * qx + qz * qz), R12 = 2.f * (qy * qz - qw * qx);
    float R20 = 2.f * (qx * qz - qw * qy), R21 = 2.f * (qy * qz + qw * qx), R22 = 1.f - 2.f * (qx * qx + qy * qy);

    float s0 = fmaxf(__expf(scl[3 * i]),     1e-7f); s0 *= s0;
    float s1 = fmaxf(__expf(scl[3 * i + 1]), 1e-7f); s1 *= s1;
    float s2 = fmaxf(__expf(scl[3 * i + 2]), 1e-7f); s2 *= s2;

    float c00 = R00 * R00 * s0 + R01 * R01 * s1 + R02 * R02 * s2;
    float c01 = R00 * R10 * s0 + R01 * R11 * s1 + R02 * R12 * s2;
    float c02 = R00 * R20 * s0 + R01 * R21 * s1 + R02 * R22 * s2;
    float c11 = R10 * R10 * s0 + R11 * R11 * s1 + R12 * R12 * s2;
    float c12 = R10 * R20 * s0 + R11 * R21 * s1 + R12 * R22 * s2;
    float c22 = R20 * R20 * s0 + R21 * R21 * s1 + R22 * R22 * s2;

    // --- projection ---
    float fx = cam[0], fy = cam[5];
    float p0 = cam[0] * x + cam[1] * y + cam[2]  * z + cam[3];
    float p1 = cam[4] * x + cam[5] * y + cam[6]  * z + cam[7];
    float dep = cam[8] * x + cam[9] * y + cam[10] * z + cam[11];
    float invd = 1.0f / (dep + 1e-7f);
    float mx = p0 * invd, my = p1 * invd;

    float J00 = fx / dep, J02 = -fx * mx / dep;
    float J11 = fy / dep, J12 = -fy * my / dep;

    float M00 = J00 * c00 + J02 * c02;
    float M01 = J00 * c01 + J02 * c12;
    float M02 = J00 * c02 + J02 * c22;
    float M10 = J11 * c01 + J12 * c02;
    float M11 = J11 * c11 + J12 * c12;
    float M12 = J11 * c12 + J12 * c22;
    float v00 = M00 * J00 + M02 * J02;
    float v01 = M01 * J11 + M02 * J12;
    float v10 = M10 * J00 + M12 * J02;
    float v11 = M11 * J11 + M12 * J12;

    float a = v00 + 1e-6f, bb = v01, cc = v10, dd = v11 + 1e-6f;
    float det = a * dd - bb * cc;
    float invdet = 1.0f / det;
    float cA = dd * invdet;
    float cB = -(bb + cc) * invdet;
    float cC = a * invdet;

    // --- SH color ---
    float vx = x - campos[0], vy = y - campos[1], vz = z - campos[2];
    float vn = rsqrtf(vx * vx + vy * vy + vz * vz);
    vx *= vn; vy *= vn; vz *= vn;
    float sh[9];
    sh[0] = 0.28209479177387814f;
    sh[1] = -0.48860251190291987f * vy;
    sh[2] =  0.48860251190291987f * vz;
    sh[3] = -0.48860251190291987f * vx;
    sh[4] =  1.0925484305920792f * vx * vy;
    sh[5] = -1.0925484305920792f * vy * vz;
    sh[6] =  0.31539156525252005f * (2.f * vz * vz - vx * vx - vy * vy);
    sh[7] = -1.0925484305920792f * vx * vz;
    sh[8] =  0.5462742152960396f * (vx * vx - vy * vy);

    float rgb[3];
    #pragma unroll
    for (int c = 0; c < 3; ++c) {
        float acc = col[3 * i + c];
        #pragma unroll
        for (int j = 0; j < 9; ++j)
            acc += sh[j] * sh_c[(i * 16 + j) * 3 + c];
        rgb[c] = sigmoidf_(acc);
    }
    float op = sigmoidf_(opa[i]);

    float* G = geom + 8 * k;
    G[0] = mx;  G[1] = my;  G[2] = cA;  G[3] = cB;
    G[4] = cC;  G[5] = op;  G[6] = 0.f; G[7] = 0.f;
    float* C = colp + 4 * k;
    C[0] = rgb[0]; C[1] = rgb[1]; C[2] = rgb[2]; C[3] = 0.f;
}

// ---- 8) render: per-wave 16-pixel tile, WMMA f32 16x16x4 color accumulation ----
__global__ __launch_bounds__(256) void render_kernel(const float* __restrict__ geom,
                                                     const float* __restrict__ colp,
                                                     float* __restrict__ out) {
    __shared__ float4 sg[KREND * 2];   // 32 KB: {mu, conic} / {conicC, op, 0, 0}
    __shared__ float4 sc[KREND];       // 16 KB: {r, g, b, 0}
    const float4* gg = (const float4*)geom;
    const float4* gc = (const float4*)colp;
    for (int i = threadIdx.x; i < KREND * 2; i += 256) sg[i] = gg[i];
    for (int i = threadIdx.x; i < KREND;     i += 256) sc[i] = gc[i];
    __syncthreads();
    const float* scf = (const float*)sc;

    const int lane = threadIdx.x & 31;
    const int wave = threadIdx.x >> 5;
    const int half = (lane >> 4) & 1;
    const int n    = lane & 15;                 // D column / pixel-row M for A
    const int nc   = (n < 3) ? n : 3;           // color channel (pad slot = 0)
    const int tileBase = blockIdx.x * 128 + wave * 16;   // 16 pixels per wave

    const int myPix = tileBase + n;
    const float fpx = (float)(myPix & (IMG_W - 1));
    const float fpy = (float)(myPix >> 8);      // IMG_W == 256

    float T = 1.0f;                             // transmittance (replicated per half)
    v8f acc = {};

    for (int c = 0; c < KREND; c += 4) {
        const int k0 = c + 2 * half;            // this half-wave's 2 gaussians
        const float4 g0a = sg[k0 * 2 + 0];      // mu.x mu.y conicA conicB
        const float4 g0b = sg[k0 * 2 + 1];      // conicC op  -   -
        const float4 g1a = sg[k0 * 2 + 2];
        const float4 g1b = sg[k0 * 2 + 3];

        float dx0 = fpx - g0a.x, dy0 = fpy - g0a.y;
        float qv0 = g0a.z * dx0 * dx0 + g0a.w * dx0 * dy0 + g0b.x * dy0 * dy0;
        float a0  = g0b.y * __expf(-0.5f * qv0);
        float dx1 = fpx - g1a.x, dy1 = fpy - g1a.y;
        float qv1 = g1a.z * dx1 * dx1 + g1a.w * dx1 * dy1 + g1b.x * dy1 * dy1;
        float a1  = g1b.y * __expf(-0.5f * qv1);

        float prodSelf  = (1.0f - a0) * (1.0f - a1);
        float prodOther = halfswap_f32(prodSelf);         // v_permlanex16_b32

        float baseT = half ? (T * prodOther) : T;         // T before this half's k's
        v2f A;
        A.x = baseT * a0;                                 // contrib of k_first
        A.y = baseT * (1.0f - a0) * a1;                   // contrib of k_first+1
        T   = T * prodSelf * prodOther;                   // advance 4 gaussians

        // B matrix: column n holds channel n of the same 2 gaussians (pad = 0)
        v2f B;
        B.x = scf[k0 * 4 + nc];
        B.y = scf[k0 * 4 + 4 + nc];

        // D(16 pix x 16 ch) += A(16x4 contrib) x B(4x16 color)
        acc = __builtin_amdgcn_wmma_f32_16x16x4_f32(
                  false, A, false, B, (short)0, acc, false, false);
    }

    // D layout: VGPR r -> pixel row (r + 8*half), column n = channel
    if (n < 3) {
        #pragma unroll
        for (int r = 0; r < 8; ++r) {
            int pix = tileBase + r + 8 * half;
            out[pix * 3 + n] = acc[r];
        }
    }
}

extern "C" void kernel_launch(void* const* d_in, const int* in_sizes, int n_in,
                              void* d_out, int out_size, void* d_ws, size_t ws_size,
                              hipStream_t stream) {
    const float* positions = (const float*)d_in[0];
    const float* scales    = (const float*)d_in[1];
    const float* rotations = (const float*)d_in[2];
    const float* colors    = (const float*)d_in[3];
    const float* opacity   = (const float*)d_in[4];
    const float* colors_sh = (const float*)d_in[5];
    const float* cam       = (const float*)d_in[6];
    const float* campos    = (const float*)d_in[7];
    const int N = in_sizes[0] / 3;
    const int K = KREND;

    char* ws = (char*)d_ws;
    size_t offKeys = 0;
    size_t offHist = ALIGN256(offKeys + (size_t)N * 4);
    size_t offCtr  = ALIGN256(offHist + (size_t)HIST_BINS * 4);
    size_t offCand = ALIGN256(offCtr + 256);
    size_t offSel  = ALIGN256(offCand + (size_t)CAND_MAX * 8);
    size_t offGeom = ALIGN256(offSel + (size_t)K * 4);
    size_t offColp = ALIGN256(offGeom + (size_t)K * 8 * 4);

    unsigned* keys = (unsigned*)(ws + offKeys);
    unsigned* hist = (unsigned*)(ws + offHist);
    unsigned* ctr  = (unsigned*)(ws + offCtr);
    unsigned long long* cand = (unsigned long long*)(ws + offCand);
    unsigned* selIdx = (unsigned*)(ws + offSel);
    float* geom = (float*)(ws + offGeom);
    float* colp = (float*)(ws + offColp);

    int gN = (N + 255) / 256;
    depth_keys_kernel<<<gN, 256, 0, stream>>>(positions, cam, keys, N);
    zero_kernel<<<(HIST_BINS + 64 + 255) / 256, 256, 0, stream>>>(hist);
    hist_kernel<<<gN, 256, 0, stream>>>(keys, hist, N);
    thresh_kernel<<<1, 32, 0, stream>>>(hist, ctr, K);
    compact_kernel<<<gN, 256, 0, stream>>>(keys, ctr, cand, N);
    sort_kernel<<<1, 512, 0, stream>>>(cand, ctr, selIdx, K);
    prep_kernel<<<(K + 255) / 256, 256, 0, stream>>>(positions, scales, rotations,
                                                     colors, opacity, colors_sh,
                                                     cam, campos, selIdx,
                                                     geom, colp, K);
    render_kernel<<<NPIX / 128, 256, 0, stream>>>(geom, colp, (float*)d_out);
}